// GraphFusionSIP_64811056496793
// MI455X (gfx1250) — compile-verified
//
#include <hip/hip_runtime.h>
#include <hip/hip_bf16.h>

typedef float v2f __attribute__((ext_vector_type(2)));
typedef float v8f __attribute__((ext_vector_type(8)));

#define IMG_D 40
#define TXT_D 20
#define FUS_D 48
#define TOT_D 144
#define NSTEP1 16   // K = 60 folded inputs + bias column, padded to 64 -> 16 steps of 4
#define NSTEP2 13   // K = 48 + bias column, padded to 52 -> 13 steps of 4

// workspace layout (floats) — weights stored in exact per-lane WMMA fragment order:
// frag f, lane l, vgpr j  ->  ws[OFF + (f*32 + l)*2 + j]
// (one coalesced global_load_b64 per fragment per lane)
#define OFF_W1F   0                        // 3 ntiles * 16 steps * 64
#define OFF_WGF   (OFF_W1F + 3*16*64)
#define OFF_W2F   (OFF_WGF + 3*16*64)      // 3 ntiles * 13 steps * 64

// ---------------- folded-weight evaluation helpers (setup kernel) ------------------
__device__ __forceinline__ float w1b_val(int k, int n,
    const float* Wiv, const float* biv, const float* Wtv, const float* btv,
    const float* Wisv, const float* bisv, const float* Wtsv, const float* btsv,
    const float* Wm1, const float* bm1)
{
    if (k < 40) {                       // image-dependent columns
        float s = 0.f;
        for (int t = 0; t < 48; ++t) s += Wm1[n * TOT_D + 48 + t] * Wtv[t * IMG_D + k];
        for (int t = 0; t < 24; ++t) s += Wm1[n * TOT_D + 96 + t] * Wisv[t * IMG_D + k];
        return s;
    } else if (k < 60) {                // text-dependent columns
        int kt = k - 40;
        float s = 0.f;
        for (int t = 0; t < 48; ++t) s += Wm1[n * TOT_D + t] * Wiv[t * TXT_D + kt];
        for (int t = 0; t < 24; ++t) s += Wm1[n * TOT_D + 120 + t] * Wtsv[t * TXT_D + kt];
        return s;
    } else if (k == 60) {               // bias column (input x[60] == 1)
        float s = bm1[n];
        for (int j = 0; j < 48; ++j)
            s += Wm1[n * TOT_D + j] * biv[j] + Wm1[n * TOT_D + 48 + j] * btv[j];
        for (int j = 0; j < 24; ++j)
            s += Wm1[n * TOT_D + 96 + j] * bisv[j] + Wm1[n * TOT_D + 120 + j] * btsv[j];
        return s;
    }
    return 0.f;                         // zero padding K = 61..63
}

__device__ __forceinline__ float wgb_val(int k, int n,
    const float* Wiv, const float* biv, const float* Wtv, const float* btv,
    const float* Wg, const float* bg)
{
    if (k < 40) {
        float s = 0.f;
        for (int t = 0; t < 48; ++t) s += Wg[n * 96 + 48 + t] * Wtv[t * IMG_D + k];
        return s;
    } else if (k < 60) {
        int kt = k - 40;
        float s = 0.f;
        for (int t = 0; t < 48; ++t) s += Wg[n * 96 + t] * Wiv[t * TXT_D + kt];
        return s;
    } else if (k == 60) {
        float s = bg[n];
        for (int j = 0; j < 48; ++j)
            s += Wg[n * 96 + j] * biv[j] + Wg[n * 96 + 48 + j] * btv[j];
        return s;
    }
    return 0.f;
}

// ---------------- setup: fold + pre-swizzle into fragment layout -------------------
__global__ void fold_weights(
    const float* __restrict__ Wiv, const float* __restrict__ biv,
    const float* __restrict__ Wtv, const float* __restrict__ btv,
    const float* __restrict__ Wisv, const float* __restrict__ bisv,
    const float* __restrict__ Wtsv, const float* __restrict__ btsv,
    const float* __restrict__ Wm1, const float* __restrict__ bm1,
    const float* __restrict__ Wm2, const float* __restrict__ bm2,
    const float* __restrict__ Wg,  const float* __restrict__ bg,
    float* __restrict__ ws)
{
    const int tid = threadIdx.x;

    // layer-1 + gate fragments: frag index f = nt*16 + s
    for (int i = tid; i < 3 * 16 * 64; i += blockDim.x) {
        int j    = i & 1;
        int l    = (i >> 1) & 31;
        int f    = i >> 6;
        int s    = f % 16, nt = f / 16;
        int n    = nt * 16 + (l & 15);
        int k    = 4 * s + ((l >> 4) << 1) + j;   // B layout: lanes0-15 K{0,1}, 16-31 K{2,3}
        ws[OFF_W1F + i] = w1b_val(k, n, Wiv, biv, Wtv, btv, Wisv, bisv,
                                  Wtsv, btsv, Wm1, bm1);
        ws[OFF_WGF + i] = wgb_val(k, n, Wiv, biv, Wtv, btv, Wg, bg);
    }

    // layer-2 fragments: frag index f = nt*13 + s; K padded 48(+bias)->52
    for (int i = tid; i < 3 * 13 * 64; i += blockDim.x) {
        int j    = i & 1;
        int l    = (i >> 1) & 31;
        int f    = i >> 6;
        int s    = f % 13, nt = f / 13;
        int n    = nt * 16 + (l & 15);
        int k    = 4 * s + ((l >> 4) << 1) + j;
        float v  = 0.f;
        if (k < 48)       v = Wm2[n * 48 + k];
        else if (k == 48) v = bm2[n];             // bias column (u[48] == 1)
        ws[OFF_W2F + i] = v;
    }
}

// ---------------- main fused kernel: 16 rows per wave, all-WMMA f32 ----------------
#define WAVES_PER_BLOCK 8
#define ROWS_PER_BLOCK  128
#define U_STRIDE        52   // u tile row stride (52 floats, gcd(52,64)=4 ok for b64 lds)

__launch_bounds__(256)
__global__ void fused_mlp(const float* __restrict__ img,
                          const float* __restrict__ txt,
                          const float* __restrict__ ws,
                          float* __restrict__ out,
                          int nrows)
{
    __shared__ float u_lds[WAVES_PER_BLOCK][16 * U_STRIDE];

    const int lane = threadIdx.x & 31;
    const int wave = threadIdx.x >> 5;
    const int nl   = lane & 15;            // N (and A-row M) index within tile
    const int hi   = lane >> 4;            // 0 = lanes 0-15, 1 = lanes 16-31
    const int koff = hi << 1;              // per-half K offset: 0 or 2
    const int row0 = blockIdx.x * ROWS_PER_BLOCK + wave * 16;
    if (row0 >= nrows) return;             // whole-wave exit keeps EXEC all-1s

    // per-lane fragment pointers: one global_load_b64 per fragment, fully coalesced
    const v2f* __restrict__ w1f = (const v2f*)(ws + OFF_W1F) + lane;
    const v2f* __restrict__ wgf = (const v2f*)(ws + OFF_WGF) + lane;
    const v2f* __restrict__ w2f = (const v2f*)(ws + OFF_W2F) + lane;

    // ---- A fragments: x tile [16 x 60] + constant-1 bias column -------------------
    v2f a[NSTEP1];
    const float* ip = img + (size_t)(row0 + nl) * IMG_D;
#pragma unroll
    for (int s = 0; s < 10; ++s)
        a[s] = *(const v2f*)(ip + 4 * s + koff);          // K = 0..39 (image)
    const float* tp = txt + (size_t)(row0 + nl) * TXT_D;
#pragma unroll
    for (int s = 0; s < 5; ++s)
        a[10 + s] = *(const v2f*)(tp + 4 * s + koff);     // K = 40..59 (text)
    a[15].x = hi ? 0.f : 1.f;                             // K=60 -> 1 (bias), 61..63 -> 0
    a[15].y = 0.f;

    // ---- layer 1 (u) and gate pre-activation share the A fragments ----------------
    v8f gacc[3];
    float* ur = u_lds[wave];
#pragma unroll
    for (int nt = 0; nt < 3; ++nt) {
        v8f c = {0.f, 0.f, 0.f, 0.f, 0.f, 0.f, 0.f, 0.f};
        v8f g = {0.f, 0.f, 0.f, 0.f, 0.f, 0.f, 0.f, 0.f};
#pragma unroll
        for (int s = 0; s < NSTEP1; ++s) {
            v2f bw = w1f[(nt * 16 + s) * 32];
            c = __builtin_amdgcn_wmma_f32_16x16x4_f32(false, a[s], false, bw,
                                                      (short)0, c, false, false);
            v2f gw = wgf[(nt * 16 + s) * 32];
            g = __builtin_amdgcn_wmma_f32_16x16x4_f32(false, a[s], false, gw,
                                                      (short)0, g, false, false);
        }
        gacc[nt] = g;
        // relu(u) -> LDS row-major [m][k]  (C/D layout: M = r + 8*hi, N = nl)
#pragma unroll
        for (int r = 0; r < 8; ++r) {
            float v = c[r] > 0.f ? c[r] : 0.f;
            ur[(r + (hi << 3)) * U_STRIDE + nt * 16 + nl] = v;
        }
    }

    // ---- A2 fragments from LDS + constant-1 bias column ---------------------------
    v2f a2[NSTEP2];
#pragma unroll
    for (int s = 0; s < 12; ++s) {
        const int k0 = 4 * s + koff;
        a2[s].x = ur[nl * U_STRIDE + k0];
        a2[s].y = ur[nl * U_STRIDE + k0 + 1];
    }
    a2[12].x = hi ? 0.f : 1.f;                            // K=48 -> 1 (bias), 49..51 -> 0
    a2[12].y = 0.f;

    // ---- layer 2 + gate + store ---------------------------------------------------
#pragma unroll
    for (int nt = 0; nt < 3; ++nt) {
        const int n = nt * 16 + nl;
        v8f c = {0.f, 0.f, 0.f, 0.f, 0.f, 0.f, 0.f, 0.f};
#pragma unroll
        for (int s = 0; s < NSTEP2; ++s) {
            v2f bw = w2f[(nt * 13 + s) * 32];
            c = __builtin_amdgcn_wmma_f32_16x16x4_f32(false, a2[s], false, bw,
                                                      (short)0, c, false, false);
        }
        v8f g = gacc[nt];
#pragma unroll
        for (int r = 0; r < 8; ++r) {
            float f  = c[r] > 0.f ? c[r] : 0.f;
            float gv = 1.0f / (1.0f + __expf(-g[r]));
            int   mm = r + (hi << 3);
            out[(size_t)(row0 + mm) * FUS_D + n] = f * gv;
        }
    }
}

extern "C" void kernel_launch(void* const* d_in, const int* in_sizes, int n_in,
                              void* d_out, int out_size, void* d_ws, size_t ws_size,
                              hipStream_t stream) {
    const float* img  = (const float*)d_in[0];
    const float* txt  = (const float*)d_in[1];
    const float* Wiv  = (const float*)d_in[6];
    const float* biv  = (const float*)d_in[7];
    const float* Wtv  = (const float*)d_in[12];
    const float* btv  = (const float*)d_in[13];
    const float* Wisv = (const float*)d_in[18];
    const float* bisv = (const float*)d_in[19];
    const float* Wtsv = (const float*)d_in[24];
    const float* btsv = (const float*)d_in[25];
    const float* Wm1  = (const float*)d_in[26];
    const float* bm1  = (const float*)d_in[27];
    const float* Wm2  = (const float*)d_in[28];
    const float* bm2  = (const float*)d_in[29];
    const float* Wg   = (const float*)d_in[30];
    const float* bg   = (const float*)d_in[31];
    float* ws = (float*)d_ws;

    fold_weights<<<1, 256, 0, stream>>>(Wiv, biv, Wtv, btv, Wisv, bisv,
                                        Wtsv, btsv, Wm1, bm1, Wm2, bm2,
                                        Wg, bg, ws);

    const int nrows = in_sizes[0] / IMG_D;                 // B = 262144
    const int grid  = (nrows + ROWS_PER_BLOCK - 1) / ROWS_PER_BLOCK;
    fused_mlp<<<grid, 256, 0, stream>>>(img, txt, ws, (float*)d_out, nrows);
}